// VAE_GNN_9981503996081
// MI455X (gfx1250) — compile-verified
//
#include <hip/hip_runtime.h>
#include <hip/hip_bf16.h>
#include <stdint.h>

// ---------------------------------------------------------------------------
// VAE-GNN forward for MI455X (gfx1250, wave32).
// Dense node GEMMs -> V_WMMA_F32_16X16X4_F32 (exact fp32 matrix pipe),
// one wave computes a 16x32 output tile (2 accumulators, shared A fetch).
// Edge softmax/aggregation -> scalar kernels with float atomics (HBM-bound).
// ---------------------------------------------------------------------------

#define NEG_SLOPE 0.01f

typedef float v2f __attribute__((ext_vector_type(2)));
typedef float v8f __attribute__((ext_vector_type(8)));

// Order-preserving float<->uint mapping for atomicMax on floats.
__device__ __forceinline__ unsigned f2ord(float f) {
  unsigned u = __float_as_uint(f);
  return (u & 0x80000000u) ? ~u : (u | 0x80000000u);
}
__device__ __forceinline__ float ord2f(unsigned u) {
  return (u & 0x80000000u) ? __uint_as_float(u & 0x7FFFFFFFu)
                           : __uint_as_float(~u);
}
// f2ord(-inf) == 0x007FFFFF : init value for the segment max.
#define ORD_NEG_INF 0x007FFFFFu

__device__ __forceinline__ v8f wmma4_f32(v2f a, v2f b, v8f c) {
  // 8-arg form: (neg_a, A, neg_b, B, c_mod, C, reuse_a, reuse_b)
  return __builtin_amdgcn_wmma_f32_16x16x4_f32(false, a, false, b, (short)0, c,
                                               false, false);
}

// ---------------------------------------------------------------------------
// GEMM: Y[M,Nout] = act( X[M,K] @ W[Nout,K]^T + bias ), all row-major fp32.
// One wave (32 lanes) computes a 16x32 tile of Y (two 16x16 WMMA tiles that
// share the A fetch), K stepped by 4 with V_WMMA_F32_16X16X4_F32.
// Requires M % 16 == 0 (M = 10000) and K % 4 == 0 (true here).
// Edge column-tiles take a branch-free masked path (clamped row pointer +
// 0/1 multiply) -- no EXEC save/restore inside the K loop.
//
// ISA operand layout (CDNA5 ISA):
//   A 16x4 fp32 : lanes 0-15 hold row m=lane, K={k,k+1}; lanes 16-31 K={k+2,k+3}
//   B 4x16 fp32 : lanes 0-15 hold col n=lane, K={k,k+1}; lanes 16-31 K={k+2,k+3}
//   C/D 16x16   : vgpr r -> row m0 + 8*half + r, col n0 + (lane&15)
// act: 0 = none, 1 = relu, 2 = leaky(0.01)
// ---------------------------------------------------------------------------
__global__ void __launch_bounds__(32)
gemm32_wmma_f32(const float* __restrict__ X, const float* __restrict__ W,
                const float* __restrict__ bias, float* __restrict__ Y,
                int M, int K, int Nout, int ldy, int act) {
  const int lane = threadIdx.x;
  const int half = lane >> 4;
  const int l    = lane & 15;
  const int m0 = blockIdx.x << 4;
  const int n0 = blockIdx.y << 5;
  const int row  = m0 + l;
  const int colA = n0 + l;
  const int colB = n0 + 16 + l;

  const float* xrow = X + (size_t)row * K;
  v8f acc0 = {0.f, 0.f, 0.f, 0.f, 0.f, 0.f, 0.f, 0.f};
  v8f acc1 = acc0;

  if (n0 + 32 <= Nout) {
    // Fast path: both column tiles fully in range, no masking at all.
    const float* wrowA = W + (size_t)colA * K;
    const float* wrowB = W + (size_t)colB * K;
    for (int k = 0; k < K; k += 4) {
      const int kk = k + (half << 1);
      v2f a, b0, b1;
      a.x  = xrow[kk];  a.y  = xrow[kk + 1];
      b0.x = wrowA[kk]; b0.y = wrowA[kk + 1];
      b1.x = wrowB[kk]; b1.y = wrowB[kk + 1];
      acc0 = wmma4_f32(a, b0, acc0);
      acc1 = wmma4_f32(a, b1, acc1);
    }
  } else {
    // Edge path: clamp row pointers (always-valid address), zero via multiply.
    const bool okA = colA < Nout, okB = colB < Nout;
    const float mA = okA ? 1.f : 0.f;
    const float mB = okB ? 1.f : 0.f;
    const float* wrowA = W + (size_t)(okA ? colA : 0) * K;
    const float* wrowB = W + (size_t)(okB ? colB : 0) * K;
    for (int k = 0; k < K; k += 4) {
      const int kk = k + (half << 1);
      v2f a, b0, b1;
      a.x  = xrow[kk];       a.y  = xrow[kk + 1];
      b0.x = wrowA[kk] * mA; b0.y = wrowA[kk + 1] * mA;
      b1.x = wrowB[kk] * mB; b1.y = wrowB[kk + 1] * mB;
      acc0 = wmma4_f32(a, b0, acc0);
      acc1 = wmma4_f32(a, b1, acc1);
    }
  }

  auto store_tile = [&](const v8f& acc, int col) {
    if (col < Nout) {
      const float bv = bias ? bias[col] : 0.f;
#pragma unroll
      for (int r = 0; r < 8; ++r) {
        const int rr = m0 + (half << 3) + r;
        float v = acc[r] + bv;
        if (act == 1)      v = fmaxf(v, 0.f);
        else if (act == 2) v = (v > 0.f) ? v : NEG_SLOPE * v;
        Y[(size_t)rr * ldy + col] = v;
      }
    }
  };
  store_tile(acc0, colA);
  store_tile(acc1, colB);
}

// ---------------------------------------------------------------------------
// Small utility kernels
// ---------------------------------------------------------------------------
__global__ void fill_u32(unsigned* __restrict__ p, size_t n, unsigned v) {
  size_t i = (size_t)blockIdx.x * blockDim.x + threadIdx.x;
  if (i < n) p[i] = v;
}

// dst[n][col0 + c] = src[n][c], c in [0,Wd)
__global__ void copy_cols(const float* __restrict__ src, int lds,
                          float* __restrict__ dst, int ldd, int col0,
                          int N, int Wd) {
  int idx = blockIdx.x * blockDim.x + threadIdx.x;
  if (idx >= N * Wd) return;
  int n = idx / Wd, c = idx % Wd;
  dst[(size_t)n * ldd + col0 + c] = src[(size_t)n * lds + c];
}

// z = mu + exp(0.5*logvar) * eps
__global__ void reparam(const float* __restrict__ mu, const float* __restrict__ lv,
                        const float* __restrict__ eps, float* __restrict__ z,
                        int total) {
  int i = blockIdx.x * blockDim.x + threadIdx.x;
  if (i < total) z[i] = mu[i] + __expf(0.5f * lv[i]) * eps[i];
}

// ---------------------------------------------------------------------------
// GAT layer kernels
// ---------------------------------------------------------------------------

// Fold the (linear) edge embedding into 3 constants per head:
//   s_e(e,h) = c0[h]*ew0 + c1[h]*ew1 + cb[h]
// where c0 = wa[h][2D:3D] . We[:,0], c1 = ... We[:,1], cb = ... be
__global__ void fold_edge_const(const float* __restrict__ wa, int Kwa, int D,
                                const float* __restrict__ We,
                                const float* __restrict__ be,
                                float* __restrict__ out, int H) {
  int h = threadIdx.x;
  if (h >= H) return;
  const float* w = wa + (size_t)h * Kwa + 2 * D;
  float c0 = 0.f, c1 = 0.f, cb = 0.f;
  for (int d = 0; d < D; ++d) {
    float wv = w[d];
    c0 += wv * We[2 * d];
    c1 += wv * We[2 * d + 1];
    cb += wv * be[d];
  }
  out[h * 3 + 0] = c0;
  out[h * 3 + 1] = c1;
  out[h * 3 + 2] = cb;
}

// Per-node attention scalars: ss[n,h] = wa[h][0:D].z[n,h,:],
//                             sd[n,h] = wa[h][D:2D].z[n,h,:]
__global__ void node_scalars(const float* __restrict__ Z,
                             const float* __restrict__ wa, int Kwa,
                             float* __restrict__ ss, float* __restrict__ sd,
                             int N, int H, int D) {
  int idx = blockIdx.x * blockDim.x + threadIdx.x;
  if (idx >= N * H) return;
  int n = idx / H, h = idx % H;
  const float* z  = Z + (size_t)n * H * D + (size_t)h * D;
  const float* w0 = wa + (size_t)h * Kwa;
  const float* w1 = w0 + D;
  float a = 0.f, b = 0.f;
  for (int d = 0; d < D; ++d) {
    float zv = z[d];
    a += w0[d] * zv;
    b += w1[d] * zv;
  }
  ss[idx] = a;
  sd[idx] = b;
}

// Attention logits (leaky-relu'd) + per-destination segment max via atomicMax.
__global__ void edge_logits(const int* __restrict__ src, const int* __restrict__ dst,
                            const float* __restrict__ ss, const float* __restrict__ sd,
                            const float* __restrict__ ew, const float* __restrict__ ec,
                            float* __restrict__ ea, unsigned* __restrict__ mbits,
                            int E, int H, int attew) {
  int e = blockIdx.x * blockDim.x + threadIdx.x;
  if (e >= E) return;
  int s = src[e], d = dst[e];
  float ew0 = 0.f, ew1 = 0.f;
  if (attew) { ew0 = ew[2 * e]; ew1 = ew[2 * e + 1]; }
  for (int h = 0; h < H; ++h) {
    float v = ss[s * H + h] + sd[d * H + h];
    if (attew) v += ec[h * 3] * ew0 + ec[h * 3 + 1] * ew1 + ec[h * 3 + 2];
    v = (v > 0.f) ? v : NEG_SLOPE * v;          // leaky relu
    ea[(size_t)e * H + h] = v;
    atomicMax(&mbits[d * H + h], f2ord(v));
  }
}

// ee = exp(e - max[dst]); segment-sum via atomicAdd. Overwrites ea with ee.
__global__ void edge_expsum(const int* __restrict__ dst, float* __restrict__ ea,
                            const unsigned* __restrict__ mbits,
                            float* __restrict__ sum, int E, int H) {
  int idx = blockIdx.x * blockDim.x + threadIdx.x;
  if (idx >= E * H) return;
  int e = idx / H, h = idx % H;
  int d = dst[e];
  float m  = ord2f(mbits[d * H + h]);
  float ee = __expf(ea[idx] - m);
  ea[idx] = ee;
  atomicAdd(&sum[d * H + h], ee);
}

// a = ee / sum[dst]  (overwrites ea with the softmax coefficient)
__global__ void edge_norm(const int* __restrict__ dst, float* __restrict__ ea,
                          const float* __restrict__ sum, int E, int H) {
  int idx = blockIdx.x * blockDim.x + threadIdx.x;
  if (idx >= E * H) return;
  int e = idx / H, h = idx % H;
  ea[idx] = ea[idx] / sum[dst[e] * H + h];
}

// agg[dst] += a * z[src]; lanes cover consecutive features of one edge
// (coalesced z loads, contiguous atomics).
__global__ void edge_aggregate(const int* __restrict__ src, const int* __restrict__ dst,
                               const float* __restrict__ ea, const float* __restrict__ Z,
                               float* __restrict__ agg, int E, int H, int D) {
  const int HD = H * D;
  int idx = blockIdx.x * blockDim.x + threadIdx.x;
  if (idx >= E * HD) return;
  int e = idx / HD, j = idx % HD;
  int h = j / D;
  float c = ea[(size_t)e * H + h];
  atomicAdd(&agg[(size_t)dst[e] * HD + j], c * Z[(size_t)src[e] * HD + j]);
}

// out[n][h*D+d] = h_in[n][d] + relu(hs[n][h*D+d] + agg[n][h*D+d])
__global__ void gat_finalize(const float* __restrict__ hin,
                             const float* __restrict__ hs,
                             const float* __restrict__ agg,
                             float* __restrict__ out,
                             int N, int H, int D, int ldo) {
  const int HD = H * D;
  int idx = blockIdx.x * blockDim.x + threadIdx.x;
  if (idx >= N * HD) return;
  int n = idx / HD, j = idx % HD;
  int d = j % D;
  float v = hs[idx] + agg[idx];
  v = fmaxf(v, 0.f);
  out[(size_t)n * ldo + j] = hin[(size_t)n * D + d] + v;
}

// ---------------------------------------------------------------------------
// Host side
// ---------------------------------------------------------------------------
static inline int cdiv(long long a, long long b) { return (int)((a + b - 1) / b); }

static void run_gat(hipStream_t st, const float* hin, int H, int D,
                    const float* ws, const float* wf, const float* wa, int Kwa,
                    bool attew, const float* We, const float* be,
                    const int* src, const int* dst, const float* ew,
                    float* Z, float* HS, float* AGG, float* SS, float* SD,
                    unsigned* MB, float* SUM, float* EA, float* EC,
                    float* out, int ldo, int N, int E) {
  const int HD = H * D;
  dim3 gg(N / 16, cdiv(HD, 32));
  // z = h @ wf^T, hs = h @ ws^T   (weights already [H*D, D] flat, head-major)
  gemm32_wmma_f32<<<gg, 32, 0, st>>>(hin, wf, nullptr, Z,  N, D, HD, HD, 0);
  gemm32_wmma_f32<<<gg, 32, 0, st>>>(hin, ws, nullptr, HS, N, D, HD, HD, 0);
  if (attew)
    fold_edge_const<<<1, H, 0, st>>>(wa, Kwa, D, We, be, EC, H);
  node_scalars<<<cdiv((long long)N * H, 256), 256, 0, st>>>(Z, wa, Kwa, SS, SD, N, H, D);
  fill_u32<<<cdiv((long long)N * H, 256), 256, 0, st>>>(MB, (size_t)N * H, ORD_NEG_INF);
  fill_u32<<<cdiv((long long)N * H, 256), 256, 0, st>>>((unsigned*)SUM, (size_t)N * H, 0u);
  fill_u32<<<cdiv((long long)N * HD, 256), 256, 0, st>>>((unsigned*)AGG, (size_t)N * HD, 0u);
  edge_logits<<<cdiv(E, 256), 256, 0, st>>>(src, dst, SS, SD, ew, EC, EA, MB, E, H, attew ? 1 : 0);
  edge_expsum<<<cdiv((long long)E * H, 256), 256, 0, st>>>(dst, EA, MB, SUM, E, H);
  edge_norm<<<cdiv((long long)E * H, 256), 256, 0, st>>>(dst, EA, SUM, E, H);
  edge_aggregate<<<cdiv((long long)E * HD, 256), 256, 0, st>>>(src, dst, EA, Z, AGG, E, H, D);
  gat_finalize<<<cdiv((long long)N * HD, 256), 256, 0, st>>>(hin, HS, AGG, out, N, H, D, ldo);
}

extern "C" void kernel_launch(void* const* d_in, const int* in_sizes, int n_in,
                              void* d_out, int out_size, void* d_ws, size_t ws_size,
                              hipStream_t stream) {
  (void)n_in; (void)out_size; (void)ws_size;
  const int N = in_sizes[0] / 24;   // feats (N, 24)
  const int E = in_sizes[5];        // src (E,)

  const float* feats = (const float*)d_in[0];
  const float* e_w   = (const float*)d_in[1];
  const float* gt    = (const float*)d_in[2];
  const float* eps   = (const float*)d_in[3];
  const int*   src   = (const int*)d_in[5];
  const int*   dst   = (const int*)d_in[6];
#define P(i) ((const float*)d_in[(i)])
  // params, dict order starting at index 7:
  //  7 emb_h_w  8 emb_h_b  9 emb_e_w 10 emb_e_b 11 emb_gt_w 12 emb_gt_b
  // 13 inp_g1_ws 14 inp_g1_wf 15 inp_g1_wa 16 inp_e2_w 17 inp_e2_b
  // 18 inp_g2_ws 19 inp_g2_wf 20 inp_g2_wa
  // 21 enc_g1_ws 22 enc_g1_wf 23 enc_g1_wa 24 enc_g2_ws 25 enc_g2_wf 26 enc_g2_wa
  // 27 mlp_lin_w 28 mlp_lin_b 29 mlp_mu_w 30 mlp_mu_b 31 mlp_lv_w 32 mlp_lv_b
  // 33 dec_g1_ws 34 dec_g1_wf 35 dec_g1_wa 36 dec_g2_ws 37 dec_g2_wf 38 dec_g2_wa
  // 39 dec0_w 40 dec0_b 41 dec1_w 42 dec1_b 43 dec2_w 44 dec2_b

  // ---- workspace carve-out (floats) ----
  float* wsf = (float*)d_ws;
  size_t off = 0;
  auto alloc = [&](size_t n) { float* p = wsf + off; off += n; return p; };
  float* Z   = alloc((size_t)N * 320);
  float* HS  = alloc((size_t)N * 320);
  float* AGG = alloc((size_t)N * 320);
  float* H1  = alloc((size_t)N * 128);
  float* H2  = alloc((size_t)N * 128);
  float* HG  = alloc((size_t)N * 128);
  float* Xc  = alloc((size_t)N * 256);
  float* MU  = alloc((size_t)N * 32);
  float* LV  = alloc((size_t)N * 32);
  float* ZL  = alloc((size_t)N * 32);
  float* HDb = alloc((size_t)N * 160);
  float* HD2 = alloc((size_t)N * 320);
  float* HD3 = alloc((size_t)N * 320);
  float* Y0  = alloc((size_t)N * 160);
  float* Y1  = alloc((size_t)N * 80);
  float* SS  = alloc((size_t)N * 2);
  float* SD  = alloc((size_t)N * 2);
  unsigned* MB = (unsigned*)alloc((size_t)N * 2);
  float* SUM = alloc((size_t)N * 2);
  float* EA  = alloc((size_t)E * 2);
  float* EC  = alloc(16);

  auto gemm = [&](const float* X, const float* W, const float* b, float* Y,
                  int K, int Nout, int ldy, int act) {
    dim3 g(N / 16, cdiv(Nout, 32));
    gemm32_wmma_f32<<<g, 32, 0, stream>>>(X, W, b, Y, N, K, Nout, ldy, act);
  };
  auto ccopy = [&](const float* s, int lds, float* d, int ldd, int c0, int Wd) {
    copy_cols<<<cdiv((long long)N * Wd, 256), 256, 0, stream>>>(s, lds, d, ldd, c0, N, Wd);
  };

  // ---- embeddings ----
  gemm(feats, P(7), P(8), H1, 24, 64, 64, 0);          // h0  [N,64]
  // ---- GNN_inp (att_ew=True) ----
  run_gat(stream, H1, 2, 64,  P(13), P(14), P(15), 192, true,  P(9),  P(10),
          src, dst, e_w, Z, HS, AGG, SS, SD, MB, SUM, EA, EC, H2, 128, N, E);
  run_gat(stream, H2, 1, 128, P(18), P(19), P(20), 384, true,  P(16), P(17),
          src, dst, e_w, Z, HS, AGG, SS, SD, MB, SUM, EA, EC, H1, 128, N, E);
  // ---- GNN_enc_gt (att_ew=False) ----
  gemm(gt, P(11), P(12), HG, 60, 64, 64, 0);           // hg0 [N,64]
  run_gat(stream, HG, 2, 64,  P(21), P(22), P(23), 128, false, nullptr, nullptr,
          src, dst, nullptr, Z, HS, AGG, SS, SD, MB, SUM, EA, EC, H2, 128, N, E);
  run_gat(stream, H2, 1, 128, P(24), P(25), P(26), 256, false, nullptr, nullptr,
          src, dst, nullptr, Z, HS, AGG, SS, SD, MB, SUM, EA, EC, HG, 128, N, E);
  // ---- MLP_Enc ----
  ccopy(H1, 128, Xc, 256, 0, 128);
  ccopy(HG, 128, Xc, 256, 128, 128);
  gemm(Xc, P(27), P(28), H2, 256, 128, 128, 2);        // leaky
  gemm(H2, P(29), P(30), MU, 128, 32, 32, 0);
  gemm(H2, P(31), P(32), LV, 128, 32, 32, 0);
  reparam<<<cdiv((long long)N * 32, 256), 256, 0, stream>>>(MU, LV, eps, ZL, N * 32);
  // ---- GNN_decoder ----
  ccopy(H1, 128, HDb, 160, 0, 128);
  ccopy(ZL, 32, HDb, 160, 128, 32);
  run_gat(stream, HDb, 2, 160, P(33), P(34), P(35), 320, false, nullptr, nullptr,
          src, dst, nullptr, Z, HS, AGG, SS, SD, MB, SUM, EA, EC, HD2, 320, N, E);
  run_gat(stream, HD2, 1, 320, P(36), P(37), P(38), 640, false, nullptr, nullptr,
          src, dst, nullptr, Z, HS, AGG, SS, SD, MB, SUM, EA, EC, HD3, 320, N, E);
  // ---- MLP_decoder ----
  gemm(HD3, P(39), P(40), Y0, 320, 160, 160, 1);
  gemm(Y0,  P(41), P(42), Y1, 160, 80, 80, 1);
  gemm(Y1,  P(43), P(44), (float*)d_out, 80, 60, 60, 0);
#undef P
}